// AI4Burgers_38096359915612
// MI455X (gfx1250) — compile-verified
//
#include <hip/hip_runtime.h>
#include <stdint.h>

// Burgers spatial operator: Lu = NU*conv(u,w1) - u_vel*conv(u,w2+w3)
// Bandwidth-bound (~3.3 FLOP/byte, ~8.6us HBM floor on MI455X).
// gfx1250 path: async global->LDS halo fill (B128 interior + B32 edges),
// s_wait_asynccnt, NT-hinted streaming for single-use u_vel/output,
// register-sliding 3x3 window for the math.

#define TILE_W 64
#define TILE_H 32
#define HALO_H (TILE_H + 2)    // 34 rows
#define LDS_STRIDE 72          // floats; 288B = multiple of 16B for B128 LDS writes
#define COL_BASE 3             // halo column c stored at index c+3 -> interior 16B-aligned
#define IMG_W 1024
#define IMG_H 1024
#define NU_CONST 0.5f

__device__ __forceinline__ void async_ld_b32(uint32_t lds_off, const float* g) {
    asm volatile("global_load_async_to_lds_b32 %0, %1, off"
                 :: "v"(lds_off), "v"(g) : "memory");
}
__device__ __forceinline__ void async_ld_b128(uint32_t lds_off, const float* g) {
    asm volatile("global_load_async_to_lds_b128 %0, %1, off"
                 :: "v"(lds_off), "v"(g) : "memory");
}
__device__ __forceinline__ void wait_asynccnt0() {
    asm volatile("s_wait_asynccnt 0" ::: "memory");
}
__device__ __forceinline__ int clampi(int v, int lo, int hi) {
    return v < lo ? lo : (v > hi ? hi : v);
}

__global__ __launch_bounds__(256)
void burgers_stencil(const float* __restrict__ u,
                     const float* __restrict__ uvel,
                     const float* __restrict__ w1,
                     const float* __restrict__ w2,
                     const float* __restrict__ w3,
                     float* __restrict__ out)
{
    __shared__ float tile[HALO_H * LDS_STRIDE];   // 9792 B

    const int tid = threadIdx.x;
    const int bx0 = blockIdx.x * TILE_W;
    const int by0 = blockIdx.y * TILE_H;
    const long long img = (long long)blockIdx.z * (long long)(IMG_W * IMG_H);
    const float* ub = u + img;

    // ---- async fill: interior 64 floats/row as B128 (16 quads x 34 rows) ----
    // global: gx = bx0 + 4q (16B aligned), LDS: (hy*72 + 4 + 4q)*4 (16B aligned)
    for (int it = tid; it < HALO_H * 16; it += 256) {
        int hy = it >> 4;
        int q  = it & 15;
        int gy = clampi(by0 + hy - 1, 0, IMG_H - 1);
        uint32_t loff = (uint32_t)(uintptr_t)(const void*)
            &tile[hy * LDS_STRIDE + COL_BASE + 1 + 4 * q];
        async_ld_b128(loff, ub + (gy * IMG_W + bx0 + 4 * q));
    }
    // ---- async fill: left/right halo columns as B32 (2 x 34) ----
    if (tid < HALO_H * 2) {
        int hy   = tid >> 1;
        int side = tid & 1;
        int gy = clampi(by0 + hy - 1, 0, IMG_H - 1);
        int gx = side ? (bx0 + TILE_W > IMG_W - 1 ? IMG_W - 1 : bx0 + TILE_W)
                      : (bx0 - 1 < 0 ? 0 : bx0 - 1);
        uint32_t loff = (uint32_t)(uintptr_t)(const void*)
            &tile[hy * LDS_STRIDE + (side ? COL_BASE + TILE_W + 1 : COL_BASE)];
        async_ld_b32(loff, ub + (gy * IMG_W + gx));
    }

    // ---- uniform weight loads (scalar path); fold NU and w2+w3 once ----
    float wa[9], wb[9];
#pragma unroll
    for (int k = 0; k < 9; ++k) {
        wa[k] = NU_CONST * w1[k];
        wb[k] = w2[k] + w3[k];
    }

    const int x  = tid & (TILE_W - 1);   // 0..63
    const int yb = (tid >> 6) * 8;       // 0,8,16,24 -> 8 rows each

    // ---- stream u_vel (single use -> non-temporal), overlapping async fill ----
    const float* uv = uvel + img;
    float uvr[8];
#pragma unroll
    for (int r = 0; r < 8; ++r)
        uvr[r] = __builtin_nontemporal_load(
            uv + (long long)(by0 + yb + r) * IMG_W + (bx0 + x));

    wait_asynccnt0();
    __syncthreads();

    // ---- sliding 3x3 window down 8 rows ----
    const float* t0 = &tile[yb * LDS_STRIDE + x + COL_BASE];
    float a0 = t0[0], a1 = t0[1], a2 = t0[2];
    t0 += LDS_STRIDE;
    float b0 = t0[0], b1 = t0[1], b2 = t0[2];

    float* ob = out + img + (long long)(by0 + yb) * IMG_W + (bx0 + x);
#pragma unroll
    for (int r = 0; r < 8; ++r) {
        t0 += LDS_STRIDE;
        float c0 = t0[0], c1 = t0[1], c2 = t0[2];

        float ca = wa[0]*a0 + wa[1]*a1 + wa[2]*a2
                 + wa[3]*b0 + wa[4]*b1 + wa[5]*b2
                 + wa[6]*c0 + wa[7]*c1 + wa[8]*c2;
        float cb = wb[0]*a0 + wb[1]*a1 + wb[2]*a2
                 + wb[3]*b0 + wb[4]*b1 + wb[5]*b2
                 + wb[6]*c0 + wb[7]*c1 + wb[8]*c2;

        __builtin_nontemporal_store(ca - uvr[r] * cb, ob);
        ob += IMG_W;

        a0 = b0; a1 = b1; a2 = b2;
        b0 = c0; b1 = c1; b2 = c2;
    }
}

extern "C" void kernel_launch(void* const* d_in, const int* in_sizes, int n_in,
                              void* d_out, int out_size, void* d_ws, size_t ws_size,
                              hipStream_t stream) {
    const float* u    = (const float*)d_in[0];
    const float* uvel = (const float*)d_in[1];
    const float* w1   = (const float*)d_in[2];
    const float* w2   = (const float*)d_in[3];
    const float* w3   = (const float*)d_in[4];
    float* out = (float*)d_out;

    const int B = in_sizes[0] / (IMG_W * IMG_H);  // 16 for the reference shapes
    dim3 grid(IMG_W / TILE_W, IMG_H / TILE_H, B);
    burgers_stencil<<<grid, 256, 0, stream>>>(u, uvel, w1, w2, w3, out);
}